// RecurrentGCN_DCRNN_15693810499715
// MI455X (gfx1250) — compile-verified
//
#include <hip/hip_runtime.h>
#include <hip/hip_bf16.h>

typedef __attribute__((ext_vector_type(2))) float v2f;
typedef __attribute__((ext_vector_type(8))) float v8f;
typedef int gvec4i __attribute__((vector_size(16)));   // matches builtin param type

#define AS1 __attribute__((address_space(1)))
#define AS3 __attribute__((address_space(3)))

#define N_FEAT  256
#define HIDDEN  32
#define IN_DIM  288                  // N_FEAT + HIDDEN
#define W1_OFF  (IN_DIM * HIDDEN)    // offset of W[1,0] inside (2,1,288,32)
#define XPITCH  68                   // 64 cols + 4 pad floats -> conflict-free frags
#define WAVES   4                    // waves per block (independent tiles)

// CDNA5 async global->LDS path (ASYNCcnt), guarded so a missing builtin
// falls back to the load+ds_store path and still compiles.
#if defined(__has_builtin)
#  if __has_builtin(__builtin_amdgcn_global_load_async_to_lds_b128) && \
      __has_builtin(__builtin_amdgcn_s_wait_asynccnt)
#    define USE_ASYNC_LDS 1
#  endif
#endif
#ifndef USE_ASYNC_LDS
#  define USE_ASYNC_LDS 0
#endif

// ---------------------------------------------------------------------------
// Kernel 1: fold W[0,0]+W[1,0] (first 256 rows only; H0==0 kills the rest) and
// pre-shuffle into V_WMMA_F32_16X16X4_F32 B-fragment order:
//   frag[((g*2 + t)*64 + s)*32 + L] = { We[4s + 2*(L>>4)    ][16t + (L&15)],
//                                       We[4s + 2*(L>>4) + 1][16t + (L&15)] }
// 8192 float2 = 64 KB, loop-invariant => WGP$/L2 resident for the main kernel.
// ---------------------------------------------------------------------------
__global__ void prep_weights(const float* __restrict__ Wz,
                             const float* __restrict__ Wh,
                             float2* __restrict__ frag) {
    int i = blockIdx.x * blockDim.x + threadIdx.x;   // 0..8191
    int L = i & 31;
    int s = (i >> 5) & 63;
    int t = (i >> 11) & 1;
    int g = (i >> 12) & 1;
    const float* W = g ? Wh : Wz;
    int n  = t * 16 + (L & 15);
    int k0 = 4 * s + 2 * (L >> 4);
    float e0 = W[k0 * HIDDEN + n]       + W[W1_OFF + k0 * HIDDEN + n];
    float e1 = W[(k0 + 1) * HIDDEN + n] + W[W1_OFF + (k0 + 1) * HIDDEN + n];
    frag[i] = make_float2(e0, e1);
}

// ---------------------------------------------------------------------------
// Kernel 2: fused  out = relu((1-sigmoid(x Wz + bz)) * tanh(x Wh + bh)) @ Wlin + blin
// One wave per 16-row tile. x is streamed exactly once; chunks of 16x64 fp32 are
// double-buffered in LDS, filled by GLOBAL_LOAD_ASYNC_TO_LDS_B128 when available.
// ---------------------------------------------------------------------------
__global__ __launch_bounds__(WAVES * 32)
void dcrnn_fused(const float* __restrict__ x,
                 const float2* __restrict__ frag,
                 const float* __restrict__ bz,
                 const float* __restrict__ bh,
                 const float* __restrict__ Wlin,
                 const float* __restrict__ blin,
                 float* __restrict__ out,
                 int nrows) {
    __shared__ float xbuf[WAVES][2][16 * XPITCH];    // 4*2*1088*4 = 34.8 KB

    const int wave = threadIdx.x >> 5;
    const int lane = threadIdx.x & 31;
    const int m    = lane & 15;                      // row (A) / col (B,C) index
    const int half = lane >> 4;
    const int kb   = 2 * half;                       // K sub-offset of this half

    const int ntiles = (nrows + 15) >> 4;
    const int tile   = blockIdx.x * WAVES + wave;    // wave-uniform
    if (tile >= ntiles) return;                      // whole-wave exit (EXEC intact)
    const int row0 = tile * 16;

    // Per-lane staging geometry: lane owns float4 slot (lr, lc4) each iteration.
    const int lr  = lane >> 4;                       // +2 rows per iteration
    const int lc4 = lane & 15;

    v8f accZ0 = {}, accZ1 = {}, accH0 = {}, accH1 = {};

    const float2* fz0 = frag;                        // (g=0, t=0)
    const float2* fz1 = frag + 2048;                 // (g=0, t=1)
    const float2* fh0 = frag + 4096;                 // (g=1, t=0)
    const float2* fh1 = frag + 6144;                 // (g=1, t=1)

    // ---- stage one 16x64 chunk into buffer b (8 x 512B wave transfers) ----
    auto stage = [&](int c, int b) {
        #pragma unroll
        for (int it = 0; it < 8; ++it) {
            int r    = it * 2 + lr;
            int grow = row0 + r;
            if (grow >= nrows) grow = nrows - 1;     // clamp (dup read, safe)
            const float* gp = x + (size_t)grow * N_FEAT + c * 64 + lc4 * 4;
            float*       lp = &xbuf[wave][b][r * XPITCH + lc4 * 4];
#if USE_ASYNC_LDS
            __builtin_amdgcn_global_load_async_to_lds_b128(
                (AS1 gvec4i*)gp, (AS3 gvec4i*)lp, /*offset=*/0, /*cpol=*/0);
#else
            *(float4*)lp = *(const float4*)gp;
#endif
        }
    };

    stage(0, 0);                                     // prefetch chunk 0
    for (int c = 0; c < 4; ++c) {                    // 4 chunks of 64 columns
        if (c < 3) stage(c + 1, (c + 1) & 1);        // overlap next copy w/ compute
#if USE_ASYNC_LDS
        // async loads complete in order: <=8 outstanding => chunk c is in LDS
        if (c < 3) __builtin_amdgcn_s_wait_asynccnt(8);
        else       __builtin_amdgcn_s_wait_asynccnt(0);
#endif
        const float* xw = xbuf[wave][c & 1];
        #pragma unroll
        for (int s = 0; s < 16; ++s) {               // 16 k-steps of K=4
            // A fragment: lane holds x[row0+m][k+kb .. +1]; pitch-68 => 0 conflicts
            v2f a = *(const v2f*)(xw + m * XPITCH + s * 4 + kb);
            int fi = (c * 16 + s) * 32 + lane;       // coalesced, cache-resident
            v2f b0 = *(const v2f*)(fz0 + fi);
            v2f b1 = *(const v2f*)(fz1 + fi);
            v2f b2 = *(const v2f*)(fh0 + fi);
            v2f b3 = *(const v2f*)(fh1 + fi);
            accZ0 = __builtin_amdgcn_wmma_f32_16x16x4_f32(false, a, false, b0,
                                                          (short)0, accZ0, false, false);
            accZ1 = __builtin_amdgcn_wmma_f32_16x16x4_f32(false, a, false, b1,
                                                          (short)0, accZ1, false, false);
            accH0 = __builtin_amdgcn_wmma_f32_16x16x4_f32(false, a, false, b2,
                                                          (short)0, accH0, false, false);
            accH1 = __builtin_amdgcn_wmma_f32_16x16x4_f32(false, a, false, b3,
                                                          (short)0, accH1, false, false);
        }
    }

    // ---- fused epilogue: gates + relu + dot with Wlin (cross-lane reduce) ----
    const float bz0 = bz[m],    bz1 = bz[m + 16];
    const float bh0 = bh[m],    bh1 = bh[m + 16];
    const float wl0 = Wlin[m],  wl1 = Wlin[m + 16];
    const float bl  = blin[0];

    #pragma unroll
    for (int r = 0; r < 8; ++r) {
        float z0 = 1.0f / (1.0f + __expf(-(accZ0[r] + bz0)));
        float z1 = 1.0f / (1.0f + __expf(-(accZ1[r] + bz1)));
        float h0 = tanhf(accH0[r] + bh0);
        float h1 = tanhf(accH1[r] + bh1);
        float v0 = (1.0f - z0) * h0;  v0 = v0 > 0.0f ? v0 : 0.0f;
        float v1 = (1.0f - z1) * h1;  v1 = v1 > 0.0f ? v1 : 0.0f;
        float v  = v0 * wl0 + v1 * wl1;
        // reduce over 16 lanes; masks < 16 keep each wave32 half independent
        #pragma unroll
        for (int msk = 1; msk < 16; msk <<= 1)
            v += __shfl_xor(v, msk, 32);
        int row = row0 + r + 8 * half;               // C layout: M = r + 8*(lane/16)
        if (m == 0 && row < nrows)
            out[row] = v + bl;
    }
}

// ---------------------------------------------------------------------------
extern "C" void kernel_launch(void* const* d_in, const int* in_sizes, int n_in,
                              void* d_out, int out_size, void* d_ws, size_t ws_size,
                              hipStream_t stream) {
    // setup_inputs order:
    // 0:x 1:edge_index(unused) 2:edge_weight(unused) 3:Wz 4:bz 5:Wr(dead) 6:br(dead)
    // 7:Wh 8:bh 9:Wlin 10:blin
    const float* x    = (const float*)d_in[0];
    const float* Wz   = (const float*)d_in[3];
    const float* bz   = (const float*)d_in[4];
    const float* Wh   = (const float*)d_in[7];
    const float* bh   = (const float*)d_in[8];
    const float* Wlin = (const float*)d_in[9];
    const float* blin = (const float*)d_in[10];
    float*  out  = (float*)d_out;
    float2* frag = (float2*)d_ws;                    // 64 KB fragment table

    int nrows  = in_sizes[0] / N_FEAT;
    int ntiles = (nrows + 15) / 16;
    int blocks = (ntiles + WAVES - 1) / WAVES;

    prep_weights<<<32, 256, 0, stream>>>(Wz, Wh, frag);
    dcrnn_fused<<<blocks, WAVES * 32, 0, stream>>>(x, frag, bz, bh, Wlin, blin, out, nrows);
}